// ReceiverGAT_38774964748932
// MI455X (gfx1250) — compile-verified
//
#include <hip/hip_runtime.h>
#include <hip/hip_bf16.h>

// ---------------------------------------------------------------------------
// ReceiverGAT on MI455X (gfx1250, wave32)
//   K1: h = x @ W  via v_wmma_f32_16x16x4_f32 (16x64 tile per wave)
//   K2: alpha_s / alpha_d per node
//   K3-K5: build dst-CSR (histogram, block-scan, scatter)
//   K6: msg_emb = message @ fc_w.T + fc_b
//   K7: per-node softmax-attention aggregation + dot with msg_emb (1 wave/node)
//   K8: per-graph log_softmax over 500 nodes
// ---------------------------------------------------------------------------

#define NNODES 50000
#define NEDGES 800000
#define NB     100
#define NPG    500
#define FDIM   64
#define HIDD   256

typedef float v2f __attribute__((ext_vector_type(2)));
typedef float v8f __attribute__((ext_vector_type(8)));

// ---- K1: h[N,64] = x[N,64] @ W[64,64] via f32 WMMA --------------------------
__global__ __launch_bounds__(256) void k_gemm_xw(const float* __restrict__ x,
                                                 const float* __restrict__ W,
                                                 float* __restrict__ h,
                                                 int nTiles) {
  int wave = blockIdx.x * (blockDim.x >> 5) + (threadIdx.x >> 5);
  if (wave >= nTiles) return;              // uniform per wave -> EXEC stays all-1
  int lane = threadIdx.x & 31;
  int half = lane >> 4;                    // 0: lanes 0-15, 1: lanes 16-31
  int lm   = lane & 15;
  int row  = wave * 16 + lm;               // A-matrix row for this lane

  v8f c0 = {}, c1 = {}, c2 = {}, c3 = {};
  const float* xrow = x + (size_t)row * FDIM;

  for (int k0 = 0; k0 < FDIM; k0 += 4) {
    int ka = k0 + 2 * half;                // K index pair held by this lane half
    v2f a;
    a.x = xrow[ka];
    a.y = xrow[ka + 1];
    const float* w0 = W + (size_t)ka * FDIM + lm;        // W[ka][n0+lm]
    const float* w1 = W + (size_t)(ka + 1) * FDIM + lm;  // W[ka+1][n0+lm]
    v2f b;
    b.x = w0[0];  b.y = w1[0];
    c0 = __builtin_amdgcn_wmma_f32_16x16x4_f32(false, a, false, b, (short)0, c0, false, false);
    b.x = w0[16]; b.y = w1[16];
    c1 = __builtin_amdgcn_wmma_f32_16x16x4_f32(false, a, false, b, (short)0, c1, false, false);
    b.x = w0[32]; b.y = w1[32];
    c2 = __builtin_amdgcn_wmma_f32_16x16x4_f32(false, a, false, b, (short)0, c2, false, false);
    b.x = w0[48]; b.y = w1[48];
    c3 = __builtin_amdgcn_wmma_f32_16x16x4_f32(false, a, false, b, (short)0, c3, false, false);
  }

  // C/D layout: VGPR j -> row (j + 8*half), column lm (+ n0)
  for (int j = 0; j < 8; ++j) {
    int r = wave * 16 + j + 8 * half;
    float* hr = h + (size_t)r * FDIM + lm;
    hr[0]  = c0[j];
    hr[16] = c1[j];
    hr[32] = c2[j];
    hr[48] = c3[j];
  }
}

// ---- K2: per-node attention logits -----------------------------------------
__global__ void k_alphas(const float* __restrict__ h,
                         const float* __restrict__ a_src,
                         const float* __restrict__ a_dst,
                         float* __restrict__ as, float* __restrict__ ad, int n) {
  int i = blockIdx.x * blockDim.x + threadIdx.x;
  if (i >= n) return;
  const float* hr = h + (size_t)i * FDIM;
  for (int hh = 0; hh < 4; ++hh) {
    float s = 0.f, d = 0.f;
    for (int dd = 0; dd < 16; ++dd) {
      float v = hr[hh * 16 + dd];
      s += v * a_src[hh * 16 + dd];
      d += v * a_dst[hh * 16 + dd];
    }
    as[i * 4 + hh] = s;
    ad[i * 4 + hh] = d;
  }
}

// ---- K3: in-degree histogram ------------------------------------------------
__global__ void k_hist(const int* __restrict__ dst, int* __restrict__ deg, int ne) {
  int e = blockIdx.x * blockDim.x + threadIdx.x;
  if (e < ne) atomicAdd(&deg[dst[e]], 1);
}

// ---- K4: single-block exclusive scan of deg -> offs[0..n] -------------------
__global__ __launch_bounds__(1024) void k_scan(const int* __restrict__ deg,
                                               int* __restrict__ offs, int n) {
  __shared__ int buf[1024];
  __shared__ int carry;
  int tid = threadIdx.x;
  if (tid == 0) carry = 0;
  __syncthreads();
  for (int base = 0; base <= n; base += 1024) {
    int idx = base + tid;
    int v = (idx < n) ? deg[idx] : 0;
    buf[tid] = v;
    __syncthreads();
    for (int d = 1; d < 1024; d <<= 1) {          // Hillis-Steele inclusive scan
      int t = (tid >= d) ? buf[tid - d] : 0;
      __syncthreads();
      buf[tid] += t;
      __syncthreads();
    }
    int incl = buf[tid];
    if (idx <= n) offs[idx] = carry + incl - v;   // exclusive
    __syncthreads();
    if (tid == 1023) carry += buf[1023];
    __syncthreads();
  }
}

// ---- K5: scatter edges into CSR by destination ------------------------------
__global__ void k_scatter(const int* __restrict__ src, const int* __restrict__ dst,
                          const int* __restrict__ offs, int* __restrict__ cursor,
                          int* __restrict__ elist, int ne) {
  int e = blockIdx.x * blockDim.x + threadIdx.x;
  if (e < ne) {
    int d = dst[e];
    int pos = offs[d] + atomicAdd(&cursor[d], 1);
    elist[pos] = src[e];
  }
}

// ---- K6: msg_emb[B,64] = message @ fc_w.T + fc_b ---------------------------
__global__ void k_msgemb(const float* __restrict__ message,
                         const float* __restrict__ fc_w,
                         const float* __restrict__ fc_b,
                         float* __restrict__ me) {
  int i = blockIdx.x * blockDim.x + threadIdx.x;
  if (i >= NB * FDIM) return;
  int b = i >> 6, c = i & 63;
  const float* m = message + (size_t)b * HIDD;
  const float* w = fc_w + (size_t)c * HIDD;
  float s = fc_b[c];
  for (int k = 0; k < HIDD; ++k) s += m[k] * w[k];
  me[i] = s;
}

// ---- K7: per-node attention softmax + aggregation + dot (one wave per node) -
__global__ __launch_bounds__(256) void k_aggregate(const float* __restrict__ h,
                                                   const float* __restrict__ as,
                                                   const float* __restrict__ ad,
                                                   const int* __restrict__ offs,
                                                   const int* __restrict__ elist,
                                                   const float* __restrict__ bias,
                                                   const float* __restrict__ me,
                                                   float* __restrict__ dots, int n) {
  int node = blockIdx.x * (blockDim.x >> 5) + (threadIdx.x >> 5);
  if (node >= n) return;
  int lane = threadIdx.x & 31;
  int b = node / NPG;
  int c0 = lane, c1 = lane + 32;           // two channels per lane
  int h0 = c0 >> 4, h1 = c1 >> 4;          // head index of each channel
  float ad0 = ad[node * 4 + h0];
  float ad1 = ad[node * 4 + h1];
  int beg = offs[node], end = offs[node + 1];

  // pass 1: per-head max (numerically stable softmax)
  float mx0 = -3.0e38f, mx1 = -3.0e38f;
  for (int j = beg; j < end; ++j) {
    int s = elist[j];
    float e0 = as[s * 4 + h0] + ad0; e0 = (e0 > 0.f) ? e0 : 0.2f * e0;
    float e1 = as[s * 4 + h1] + ad1; e1 = (e1 > 0.f) ? e1 : 0.2f * e1;
    mx0 = fmaxf(mx0, e0);
    mx1 = fmaxf(mx1, e1);
  }

  // pass 2: exp-sum and weighted aggregation of h[src]
  float den0 = 0.f, den1 = 0.f, acc0 = 0.f, acc1 = 0.f;
  for (int j = beg; j < end; ++j) {
    int s = elist[j];
    float e0 = as[s * 4 + h0] + ad0; e0 = (e0 > 0.f) ? e0 : 0.2f * e0;
    float e1 = as[s * 4 + h1] + ad1; e1 = (e1 > 0.f) ? e1 : 0.2f * e1;
    float x0 = __expf(e0 - mx0), x1 = __expf(e1 - mx1);
    den0 += x0; den1 += x1;
    const float* hs = h + (size_t)s * FDIM;
    acc0 += x0 * hs[c0];
    acc1 += x1 * hs[c1];
  }

  float o0 = (end > beg) ? acc0 / den0 : 0.f;
  float o1 = (end > beg) ? acc1 / den1 : 0.f;
  o0 += bias[c0];
  o1 += bias[c1];

  float p = o0 * me[b * FDIM + c0] + o1 * me[b * FDIM + c1];
  for (int o = 16; o > 0; o >>= 1) p += __shfl_xor(p, o, 32);
  if (lane == 0) dots[node] = p;
}

// ---- K8: log_softmax over each graph's 500 nodes ----------------------------
__global__ __launch_bounds__(256) void k_logsoftmax(const float* __restrict__ dots,
                                                    float* __restrict__ out) {
  __shared__ float red[8];
  int b = blockIdx.x;
  const float* v = dots + (size_t)b * NPG;
  int tid = threadIdx.x;
  int wid = tid >> 5, lane = tid & 31;

  float mx = -3.0e38f;
  for (int i = tid; i < NPG; i += blockDim.x) mx = fmaxf(mx, v[i]);
  for (int o = 16; o > 0; o >>= 1) mx = fmaxf(mx, __shfl_xor(mx, o, 32));
  if (lane == 0) red[wid] = mx;
  __syncthreads();
  float m2 = red[0];
  for (int i = 1; i < 8; ++i) m2 = fmaxf(m2, red[i]);
  __syncthreads();

  float sm = 0.f;
  for (int i = tid; i < NPG; i += blockDim.x) sm += __expf(v[i] - m2);
  for (int o = 16; o > 0; o >>= 1) sm += __shfl_xor(sm, o, 32);
  if (lane == 0) red[wid] = sm;
  __syncthreads();
  float tot = 0.f;
  for (int i = 0; i < 8; ++i) tot += red[i];
  float lg = __logf(tot);

  for (int i = tid; i < NPG; i += blockDim.x)
    out[(size_t)b * NPG + i] = v[i] - m2 - lg;
}

// ---------------------------------------------------------------------------
extern "C" void kernel_launch(void* const* d_in, const int* in_sizes, int n_in,
                              void* d_out, int out_size, void* d_ws, size_t ws_size,
                              hipStream_t stream) {
  (void)in_sizes; (void)n_in; (void)out_size; (void)ws_size;

  const float* message = (const float*)d_in[0];
  const float* x       = (const float*)d_in[1];
  const int*   esrc    = (const int*)  d_in[2];
  const int*   edst    = (const int*)  d_in[3];
  const float* W       = (const float*)d_in[4];
  const float* a_src   = (const float*)d_in[5];
  const float* a_dst   = (const float*)d_in[6];
  const float* bias    = (const float*)d_in[7];
  const float* fc_w    = (const float*)d_in[8];
  const float* fc_b    = (const float*)d_in[9];
  float* out = (float*)d_out;

  // carve workspace (256B aligned chunks)
  size_t off = 0;
  char* base = (char*)d_ws;
  auto carve = [&](size_t bytes) -> void* {
    void* p = base + off;
    off += (bytes + 255) & ~(size_t)255;
    return p;
  };
  float* h       = (float*)carve((size_t)NNODES * FDIM * 4);   // 12.8 MB
  float* as      = (float*)carve((size_t)NNODES * 4 * 4);
  float* ad      = (float*)carve((size_t)NNODES * 4 * 4);
  int*   deg     = (int*)  carve((size_t)NNODES * 4);
  int*   offs    = (int*)  carve((size_t)(NNODES + 1) * 4);
  int*   cursor  = (int*)  carve((size_t)NNODES * 4);
  int*   elist   = (int*)  carve((size_t)NEDGES * 4);          // 3.2 MB
  float* me      = (float*)carve((size_t)NB * FDIM * 4);
  float* dots    = (float*)carve((size_t)NNODES * 4);

  hipMemsetAsync(deg,    0, (size_t)NNODES * 4, stream);
  hipMemsetAsync(cursor, 0, (size_t)NNODES * 4, stream);

  // K1: GEMM (WMMA f32 16x16x4), one 16-row tile per wave, 8 waves per block
  int nTiles = NNODES / 16;                         // 3125
  k_gemm_xw<<<(nTiles + 7) / 8, 256, 0, stream>>>(x, W, h, nTiles);

  // K2: alphas
  k_alphas<<<(NNODES + 255) / 256, 256, 0, stream>>>(h, a_src, a_dst, as, ad, NNODES);

  // K3-K5: CSR build
  k_hist<<<(NEDGES + 255) / 256, 256, 0, stream>>>(edst, deg, NEDGES);
  k_scan<<<1, 1024, 0, stream>>>(deg, offs, NNODES);
  k_scatter<<<(NEDGES + 255) / 256, 256, 0, stream>>>(esrc, edst, offs, cursor, elist, NEDGES);

  // K6: msg_emb
  k_msgemb<<<(NB * FDIM + 255) / 256, 256, 0, stream>>>(message, fc_w, fc_b, me);

  // K7: aggregation + dot (one wave per node, 8 waves per block)
  k_aggregate<<<(NNODES + 7) / 8, 256, 0, stream>>>(h, as, ad, offs, elist, bias, me, dots, NNODES);

  // K8: per-graph log_softmax
  k_logsoftmax<<<NB, 256, 0, stream>>>(dots, out);
}